// Renderer_87325275062614
// MI455X (gfx1250) — compile-verified
//
#include <hip/hip_runtime.h>

#define IMAGE    128
#define N_LINES  32
#define N_CURVES 64          // N_LINES * N_SEGMENTS
#define SEG_PAD  16          // 9 real segments padded to 16 for WMMA M-dim
#define N_SEG    9           // N_SAMPLES - 1
#define SIGMA2   0.05f
#define EPSF     1e-10f
#define BIGF     1e30f

typedef float v2f __attribute__((ext_vector_type(2)));
typedef float v8f __attribute__((ext_vector_type(8)));

__device__ __forceinline__ float clamp01(float v) { return fminf(fmaxf(v, 0.0f), 1.0f); }

__global__ __launch_bounds__(256)
void render_curves_wmma(const float* __restrict__ position,   // (4,32,10)
                        const float* __restrict__ radius,     // (4,32,1)
                        const float* __restrict__ color,      // (4,32,3)
                        const float* __restrict__ background, // (4,3)
                        float* __restrict__ out)              // (4,3,128,128)
{
    // A-fragment tables, one 4-float K-row per (curve, padded segment):
    //  sS1: [vy,vx,-a.v,0]*inv(vv+eps)  -> WMMA gives unclamped tp
    //  sS2: [-2ay,-2ax,|a|^2,1]         -> WMMA gives |d|^2
    //  sS3: [0,0,vv,0]                  -> WMMA broadcasts vv per segment row
    __align__(16) __shared__ float sS1[N_CURVES][SEG_PAD][4];
    __align__(16) __shared__ float sS2[N_CURVES][SEG_PAD][4];
    __align__(16) __shared__ float sS3[N_CURVES][SEG_PAD][4];
    __align__(16) __shared__ float sMeta[N_CURVES][4];        // {1/sigma, cR, cG, cB}

    const int b   = blockIdx.x / IMAGE;   // batch
    const int y   = blockIdx.x % IMAGE;   // image row handled by this block
    const int tid = threadIdx.x;

    // ---------------- setup phase: build per-batch segment tables ----------------
    for (int e = tid; e < N_CURVES * SEG_PAD; e += 256) {
        const int curve = e >> 4;
        const int seg   = e & 15;
        if (seg < N_SEG) {
            const int line = curve >> 1;
            const float* P = position + (b * N_LINES + line) * 10;
            float c0y, c0x, c1y, c1x, c2y, c2x;
            if ((curve & 1) == 0) {               // first segment: points 0,1,2
                c0y = P[0]; c0x = P[1];
                c1y = P[2]; c1x = P[3];
                c2y = P[4]; c2x = P[5];
            } else {                               // second: P2, 2*P2-P3, P4 (reflection)
                c0y = P[4];               c0x = P[5];
                c1y = 2.0f * P[4] - P[6]; c1x = 2.0f * P[5] - P[7];
                c2y = P[8];               c2x = P[9];
            }
            const float t0 = (float)seg       * (1.0f / 9.0f);
            const float t1 = (float)(seg + 1) * (1.0f / 9.0f);
            const float u0 = 1.0f - t0, u1 = 1.0f - t1;
            const float ay = u0*u0*c0y + 2.0f*t0*u0*c1y + t0*t0*c2y;
            const float ax = u0*u0*c0x + 2.0f*t0*u0*c1x + t0*t0*c2x;
            const float ey = u1*u1*c0y + 2.0f*t1*u1*c1y + t1*t1*c2y;
            const float ex = u1*u1*c0x + 2.0f*t1*u1*c1x + t1*t1*c2x;
            const float vy = ey - ay, vx = ex - ax;
            const float vv = vy*vy + vx*vx;
            const float inv = 1.0f / (vv + EPSF);
            sS1[curve][seg][0] = vy * inv;
            sS1[curve][seg][1] = vx * inv;
            sS1[curve][seg][2] = -(ay*vy + ax*vx) * inv;
            sS1[curve][seg][3] = 0.0f;
            sS2[curve][seg][0] = -2.0f * ay;
            sS2[curve][seg][1] = -2.0f * ax;
            sS2[curve][seg][2] = ay*ay + ax*ax;
            sS2[curve][seg][3] = 1.0f;
            sS3[curve][seg][0] = 0.0f; sS3[curve][seg][1] = 0.0f;
            sS3[curve][seg][2] = vv;   sS3[curve][seg][3] = 0.0f;
        } else {  // padded segment -> edt2 = BIG, never selected by min
            sS1[curve][seg][0] = 0.0f; sS1[curve][seg][1] = 0.0f;
            sS1[curve][seg][2] = 0.0f; sS1[curve][seg][3] = 0.0f;
            sS2[curve][seg][0] = 0.0f; sS2[curve][seg][1] = 0.0f;
            sS2[curve][seg][2] = BIGF; sS2[curve][seg][3] = 0.0f;
            sS3[curve][seg][0] = 0.0f; sS3[curve][seg][1] = 0.0f;
            sS3[curve][seg][2] = 0.0f; sS3[curve][seg][3] = 0.0f;
        }
    }
    if (tid < N_CURVES) {
        const int line = tid >> 1;
        const float r  = radius[b * N_LINES + line];
        const float sg = r * r * SIGMA2;
        sMeta[tid][0] = 1.0f / sg;
        sMeta[tid][1] = color[(b * N_LINES + line) * 3 + 0];
        sMeta[tid][2] = color[(b * N_LINES + line) * 3 + 1];
        sMeta[tid][3] = color[(b * N_LINES + line) * 3 + 2];
    }
    __syncthreads();

    // ---------------- render phase: 1 wave = 16-pixel tile of this row ----------------
    const int wave = tid >> 5;          // 0..7
    const int lane = tid & 31;
    const int colp = lane & 15;         // pixel column within tile (B N-index)
    const int x    = wave * 16 + colp;
    const float sc = 2.0f / 127.0f;
    const float gy = -1.0f + (float)y * sc;
    const float gx = -1.0f + (float)x * sc;
    const bool  hi = lane >= 16;

    // B fragment (4x16, K rows striped across lane halves like A):
    //   VGPR0: K0 = gy (lanes 0-15) | K2 = 1.0  (lanes 16-31)
    //   VGPR1: K1 = gx (lanes 0-15) | K3 = |p|^2 (lanes 16-31)
    v2f Bf;
    Bf.x = hi ? 1.0f : gy;
    Bf.y = hi ? (gy * gy + gx * gx) : gx;

    // A fragment fetch: lane supplies segment (lane&15); K offset 0 (low half) / 2 (high half)
    const int segSel = lane & 15;
    const int kOff   = hi ? 2 : 0;

    float accR = 0.0f, accG = 0.0f, accB = 0.0f;
    float w = 1.0f;                     // exclusive transmittance

    for (int curve = 0; curve < N_CURVES; ++curve) {
        const v2f A1 = *(const v2f*)&sS1[curve][segSel][kOff];
        const v2f A2 = *(const v2f*)&sS2[curve][segSel][kOff];
        const v2f A3 = *(const v2f*)&sS3[curve][segSel][kOff];

        v8f U  = {}; // unclamped tp   per (segment row, pixel col)
        v8f S2 = {}; // |d|^2
        v8f VV = {}; // vv
        U  = __builtin_amdgcn_wmma_f32_16x16x4_f32(false, A1, false, Bf, (short)0, U,  false, false);
        S2 = __builtin_amdgcn_wmma_f32_16x16x4_f32(false, A2, false, Bf, (short)0, S2, false, false);
        VV = __builtin_amdgcn_wmma_f32_16x16x4_f32(false, A3, false, Bf, (short)0, VV, false, false);

        // edt2 = |d|^2 - vv*tp*(2u - tp),  tp = clamp(u,0,1); min over this lane's 8 segments
        float m = BIGF;
#pragma unroll
        for (int i = 0; i < 8; ++i) {
            const float u  = U[i];
            const float tp = clamp01(u);
            const float e2 = S2[i] - VV[i] * (tp * (2.0f * u - tp));
            m = fminf(m, e2);
        }
        // combine low-half (segments 0-7) with high-half (segments 8-15)
        m = fminf(m, __shfl_xor(m, 16, 32));

        const float invsig = sMeta[curve][0];
        const float z = m * invsig;
        const float I = __expf(-(z * z));          // exp(-(edt2/sigma)^2)
        const float wI = w * I;
        accR = fmaf(sMeta[curve][1], wI, accR);
        accG = fmaf(sMeta[curve][2], wI, accG);
        accB = fmaf(sMeta[curve][3], wI, accB);
        w = w * (1.0f - I + EPSF);
    }

    // background composited with final transmittance; both lane halves hold identical
    // results -> only low half writes.
    if (!hi) {
        const float bg0 = background[b * 3 + 0];
        const float bg1 = background[b * 3 + 1];
        const float bg2 = background[b * 3 + 2];
        const int base = ((b * 3) * IMAGE + y) * IMAGE + x;
        out[base]                     = clamp01(fmaf(bg0, w, accR));
        out[base + IMAGE * IMAGE]     = clamp01(fmaf(bg1, w, accG));
        out[base + 2 * IMAGE * IMAGE] = clamp01(fmaf(bg2, w, accB));
    }
}

extern "C" void kernel_launch(void* const* d_in, const int* in_sizes, int n_in,
                              void* d_out, int out_size, void* d_ws, size_t ws_size,
                              hipStream_t stream) {
    (void)in_sizes; (void)n_in; (void)out_size; (void)d_ws; (void)ws_size;
    const float* position   = (const float*)d_in[0];
    const float* radius     = (const float*)d_in[1];
    const float* color      = (const float*)d_in[2];
    const float* background = (const float*)d_in[3];
    float* out = (float*)d_out;
    // one block per (batch, image row): 4*128 = 512 blocks, 256 threads (8 waves)
    render_curves_wmma<<<4 * IMAGE, 256, 0, stream>>>(position, radius, color, background, out);
}